// EmbeddingLoss_4638564680071
// MI455X (gfx1250) — compile-verified
//
#include <hip/hip_runtime.h>
#include <hip/hip_bf16.h>

// ---------------------------------------------------------------------------
// Problem constants (from the reference)
// ---------------------------------------------------------------------------
#define BATCH   8
#define CHANS   64
#define HW      76800           // 240*320
#define KCLUS   10
#define KP      16              // K padded to 16 for WMMA tiles
#define ALPHA_C 0.02f
#define DELTA_C 0.5f
#define MINW    50.0f

// workspace layout (float offsets)
#define SUMS_OFF    0                       // B*C*KP      = 8192
#define COUNTS_OFF  8192                    // B*KP        = 128
#define MEANS_OFF   8320                    // B*C*KP      = 8192
#define NK_OFF      16512                   // B*KP        = 128
#define SCAL_OFF    16640                   // [0]=intra_sum [1]=inter_sum
#define ZERO_N      16642                   // floats to zero each call
#define INTRAD_OFF  16642                   // B*HW        = 614400

typedef float v2f __attribute__((ext_vector_type(2)));
typedef float v8f __attribute__((ext_vector_type(8)));

// ---------------------------------------------------------------------------
// 0) zero the accumulator region of the workspace
// ---------------------------------------------------------------------------
__global__ void el_init_kernel(float* ws, int n) {
    int i = blockIdx.x * blockDim.x + threadIdx.x;
    for (; i < n; i += gridDim.x * blockDim.x) ws[i] = 0.0f;
}

// ---------------------------------------------------------------------------
// 1) counts[b,k] = #pixels with cm==k   (LDS histogram per block)
//    grid = B*HW/256 = 2400 blocks of 256 (300 blocks per batch)
// ---------------------------------------------------------------------------
__global__ void el_count_kernel(const int* __restrict__ cm, float* __restrict__ counts) {
    __shared__ int h[KP];
    int tid = threadIdx.x;
    if (tid < KP) h[tid] = 0;
    __syncthreads();
    int b = blockIdx.x / (HW / 256);
    int p = (blockIdx.x % (HW / 256)) * 256 + tid;
    int c = cm[(size_t)b * HW + p];
    atomicAdd(&h[c], 1);
    __syncthreads();
    if (tid < KP && h[tid] != 0)
        atomicAdd(&counts[b * KP + tid], (float)h[tid]);
}

// ---------------------------------------------------------------------------
// 2) sums[b,c,k] = sum_p x[b,c,p] * onehot[p,k]  via V_WMMA_F32_16X16X4_F32
//    One wave handles (batch, 16-channel group, 1024-pixel chunk).
//    A (16x4 f32): lane<16 -> M=lane, K={0,1}; lane>=16 -> M=lane-16, K={2,3}
//    B (4x16 f32): lane<16 -> N=lane, K={0,1}; lane>=16 -> N=lane-16, K={2,3}
//    D (16x16):    row = j + 8*(lane>=16), col = lane&15
//    Exact in f32 since B entries are 0.0/1.0.
// ---------------------------------------------------------------------------
#define CHUNK_PIX 1024
#define GROUPS    (CHANS / 16)                 // 4
#define CHUNKS    (HW / CHUNK_PIX)             // 75
#define WAVES_PB  (GROUPS * CHUNKS)            // 300 waves per batch

__global__ void el_sums_wmma_kernel(const float* __restrict__ x,
                                    const int*   __restrict__ cm,
                                    float*       __restrict__ sums) {
    int lane  = threadIdx.x & 31;
    int gwave = blockIdx.x * (blockDim.x >> 5) + (threadIdx.x >> 5);
    int b     = gwave / WAVES_PB;
    int r     = gwave % WAVES_PB;
    int grp   = r / CHUNKS;
    int chunk = r % CHUNKS;

    int m     = lane & 15;                 // channel row within group / cluster col
    int khalf = (lane >> 4) ? 2 : 0;       // pixel sub-index this half-wave covers
    int cbase = grp * 16;
    int p0    = chunk * CHUNK_PIX;

    const float* xrow = x  + ((size_t)b * CHANS + cbase + m) * HW;
    const int*   crow = cm + (size_t)b * HW;

    v8f acc0 = {}, acc1 = {};
    for (int p = p0; p < p0 + CHUNK_PIX; p += 8) {
        // ---- tile at p ----
        float2 av = *(const float2*)(xrow + p + khalf);           // 8B aligned
        int2   cv = *(const int2*)(crow + p + khalf);
        v2f a; a.x = av.x;                       a.y = av.y;
        v2f bb; bb.x = (cv.x == m) ? 1.0f : 0.0f; bb.y = (cv.y == m) ? 1.0f : 0.0f;
        acc0 = __builtin_amdgcn_wmma_f32_16x16x4_f32(false, a, false, bb,
                                                     (short)0, acc0, false, false);
        // ---- tile at p+4 (independent accumulator for ILP) ----
        float2 av2 = *(const float2*)(xrow + p + 4 + khalf);
        int2   cv2 = *(const int2*)(crow + p + 4 + khalf);
        v2f a2; a2.x = av2.x;                        a2.y = av2.y;
        v2f b2; b2.x = (cv2.x == m) ? 1.0f : 0.0f;   b2.y = (cv2.y == m) ? 1.0f : 0.0f;
        acc1 = __builtin_amdgcn_wmma_f32_16x16x4_f32(false, a2, false, b2,
                                                     (short)0, acc1, false, false);
    }

    int rowoff = (lane >> 4) ? 8 : 0;
    #pragma unroll
    for (int j = 0; j < 8; ++j) {
        int ch = cbase + j + rowoff;
        atomicAdd(&sums[(size_t)b * CHANS * KP + ch * KP + (lane & 15)],
                  acc0[j] + acc1[j]);
    }
}

// ---------------------------------------------------------------------------
// 3) means = normalize(sums/counts); Gram G=meansT*means via WMMA -> inter hinge
//    one block (256 thr) per batch
// ---------------------------------------------------------------------------
__global__ void el_means_gram_kernel(const float* __restrict__ sums,
                                     const float* __restrict__ counts,
                                     float*       __restrict__ means,
                                     float*       __restrict__ scal) {
    __shared__ float ms[CHANS * KP];     // [c][k]
    __shared__ float nrm[KP];
    int tid = threadIdx.x;
    int b   = blockIdx.x;

    for (int idx = tid; idx < CHANS * KP; idx += 256) {
        int k = idx & 15;
        ms[idx] = sums[(size_t)b * CHANS * KP + idx] / (counts[b * KP + k] + 1e-10f);
    }
    __syncthreads();
    if (tid < KP) {
        float s = 0.0f;
        for (int c = 0; c < CHANS; ++c) { float v = ms[c * KP + tid]; s += v * v; }
        nrm[tid] = fmaxf(sqrtf(s), 1e-12f);
    }
    __syncthreads();
    for (int idx = tid; idx < CHANS * KP; idx += 256) {
        float v = ms[idx] / nrm[idx & 15];
        ms[idx] = v;
        means[(size_t)b * CHANS * KP + idx] = v;
    }
    __syncthreads();

    if (tid < 32) {                      // wave 0, EXEC all ones
        int lane  = tid;
        int m     = lane & 15;           // Gram row for A == Gram col for B
        int khalf = (lane >> 4) ? 2 : 0; // channel sub-index
        v8f g = {};
        for (int c0 = 0; c0 < CHANS; c0 += 4) {
            v2f a;
            a.x = ms[(c0 + khalf)     * KP + m];
            a.y = ms[(c0 + khalf + 1) * KP + m];
            // for a Gram matrix the A and B lane-fragments coincide
            g = __builtin_amdgcn_wmma_f32_16x16x4_f32(false, a, false, a,
                                                      (short)0, g, false, false);
        }
        int rowoff = (lane >> 4) ? 8 : 0;
        int col    = lane & 15;
        float s = 0.0f;
        #pragma unroll
        for (int j = 0; j < 8; ++j) {
            int row = j + rowoff;
            if (row < KCLUS && col < KCLUS && row != col) {
                float d = 0.5f * (1.0f - g[j]);
                float h = fmaxf(DELTA_C - d, 0.0f);
                s += h * h;
            }
        }
        atomicAdd(&scal[1], s);
    }
}

// ---------------------------------------------------------------------------
// 4) per-pixel intra_d + indicator histogram N_k
//    grid = 2400 blocks of 256 (300 per batch); means staged in LDS
// ---------------------------------------------------------------------------
__global__ void el_intra_kernel(const float* __restrict__ x,
                                const int*   __restrict__ cm,
                                const float* __restrict__ means,
                                float*       __restrict__ intrad,
                                float*       __restrict__ nk) {
    __shared__ float ms[CHANS * KP];
    __shared__ int   h[KP];
    int tid = threadIdx.x;
    int b   = blockIdx.x / (HW / 256);
    int p   = (blockIdx.x % (HW / 256)) * 256 + tid;

    for (int idx = tid; idx < CHANS * KP; idx += 256)
        ms[idx] = means[(size_t)b * CHANS * KP + idx];
    if (tid < KP) h[tid] = 0;
    __syncthreads();

    int cmp = cm[(size_t)b * HW + p];
    const float* xb = x + (size_t)b * CHANS * HW + p;
    float dot = 0.0f;
    #pragma unroll 8
    for (int c = 0; c < CHANS; ++c) {
        if ((c & 7) == 0 && c + 8 < CHANS)
            __builtin_prefetch(xb + (size_t)(c + 8) * HW, 0, 1);   // global_prefetch_b8
        dot += xb[(size_t)c * HW] * ms[c * KP + cmp];
    }
    float d = 0.5f * (1.0f - dot);
    intrad[(size_t)b * HW + p] = d;
    if (d > ALPHA_C) atomicAdd(&h[cmp], 1);
    __syncthreads();
    if (tid < KP && h[tid] != 0)
        atomicAdd(&nk[b * KP + tid], (float)h[tid]);
}

// ---------------------------------------------------------------------------
// 5) sum of intra_d^2 / (max(N_k[cm],50)*K)   with block reduction
// ---------------------------------------------------------------------------
__global__ void el_intra_loss_kernel(const int*   __restrict__ cm,
                                     const float* __restrict__ intrad,
                                     const float* __restrict__ nk,
                                     float*       __restrict__ scal) {
    __shared__ float red[256];
    int tid = threadIdx.x;
    int b   = blockIdx.x / (HW / 256);
    int p   = (blockIdx.x % (HW / 256)) * 256 + tid;
    int cmp = cm[(size_t)b * HW + p];
    float w = fmaxf(nk[b * KP + cmp], MINW) * (float)KCLUS;
    float d = intrad[(size_t)b * HW + p];
    red[tid] = d * d / w;
    __syncthreads();
    for (int s = 128; s > 0; s >>= 1) {
        if (tid < s) red[tid] += red[tid + s];
        __syncthreads();
    }
    if (tid == 0) atomicAdd(&scal[0], red[0]);
}

// ---------------------------------------------------------------------------
// 6) finalize -> out[0]=loss, out[1]=intra_loss, out[2]=inter_loss
// ---------------------------------------------------------------------------
__global__ void el_final_kernel(const float* __restrict__ nk,
                                const float* __restrict__ scal,
                                float*       __restrict__ out) {
    float tot = 0.0f;
    for (int i = 0; i < BATCH * KP; ++i) tot += nk[i];
    float intra = (tot > 0.0f) ? (scal[0] / (float)BATCH) : 0.0f;   // LAMBDA_INTRA=1
    float inter = scal[1] / ((float)(KCLUS * (KCLUS - 1) / 2) * (float)BATCH);
    out[0] = intra + inter;
    out[1] = intra;
    out[2] = inter;
}

// ---------------------------------------------------------------------------
extern "C" void kernel_launch(void* const* d_in, const int* in_sizes, int n_in,
                              void* d_out, int out_size, void* d_ws, size_t ws_size,
                              hipStream_t stream) {
    const float* x  = (const float*)d_in[0];
    const int*   cm = (const int*)d_in[1];
    float* ws  = (float*)d_ws;
    float* out = (float*)d_out;

    float* sums   = ws + SUMS_OFF;
    float* counts = ws + COUNTS_OFF;
    float* means  = ws + MEANS_OFF;
    float* nk     = ws + NK_OFF;
    float* scal   = ws + SCAL_OFF;
    float* intrad = ws + INTRAD_OFF;

    el_init_kernel<<<(ZERO_N + 255) / 256, 256, 0, stream>>>(ws, ZERO_N);

    int pix_blocks = BATCH * (HW / 256);               // 2400
    el_count_kernel<<<pix_blocks, 256, 0, stream>>>(cm, counts);

    int total_waves = BATCH * WAVES_PB;                // 2400 waves
    el_sums_wmma_kernel<<<total_waves / 8, 256, 0, stream>>>(x, cm, sums);

    el_means_gram_kernel<<<BATCH, 256, 0, stream>>>(sums, counts, means, scal);

    el_intra_kernel<<<pix_blocks, 256, 0, stream>>>(x, cm, means, intrad, nk);

    el_intra_loss_kernel<<<pix_blocks, 256, 0, stream>>>(cm, intrad, nk, scal);

    el_final_kernel<<<1, 1, 0, stream>>>(nk, scal, out);
}